// DotProductAttention_50663434223778
// MI455X (gfx1250) — compile-verified
//
#include <hip/hip_runtime.h>

// B=16, P=64, L=256, D=128  -- fused  (Q@K^T)*mask/sqrt(D)*V  (elementwise with V)
// Memory-bound kernel (~768 MB HBM traffic, ~33us floor @ 23.3 TB/s), fp32 WMMA
// on the matrix pipe for the 17.2 GFLOP of GEMM work.

typedef __attribute__((ext_vector_type(2))) float v2f;
typedef __attribute__((ext_vector_type(8))) float v8f;

#define B_ 16
#define P_ 64
#define L_ 256
#define D_ 128
#define INV_SQRT_D 0.08838834764831845f  // 1/sqrt(128)

// One wave32 computes one 16x16 tile of out[bp] = (Q Kt^T)*s*V.
// 8 waves per 256-thread block; grid covers all B*P*(L/16)*(L/16) tiles exactly.
__global__ __launch_bounds__(256) void attn_qk_mask_v_kernel(
    const float* __restrict__ Q,    // [B*P, L, D]
    const float* __restrict__ Kt,   // [B*P, L, D]
    const float* __restrict__ V,    // [B*P, L, L]
    const float* __restrict__ Msk,  // [B*P]
    float* __restrict__ Out)        // [B*P, L, L]
{
    const int lane      = threadIdx.x & 31;
    const int waveInBlk = threadIdx.x >> 5;
    const int wave      = blockIdx.x * 8 + waveInBlk;

    const int bp   = wave >> 8;        // 0 .. B*P-1
    const int tile = wave & 255;       // 16x16 grid of 16x16 tiles
    const int m0   = (tile >> 4) << 4; // tile row origin
    const int n0   = (tile & 15) << 4; // tile col origin

    const int lo16 = lane & 15;        // M (for A) / N (for B,C,D)
    const int hi   = lane >> 4;        // selects K=2,3 half for A/B operands

    // A operand: lane holds Q[m0+lo16][kk + 2*hi + {0,1}]  -> contiguous float2
    const float* qp = Q  + ((size_t)bp * L_ + (m0 + lo16)) * (size_t)D_ + 2 * hi;
    // B operand: lane holds Kt[n0+lo16][kk + 2*hi + {0,1}] -> contiguous float2
    const float* kp = Kt + ((size_t)bp * L_ + (n0 + lo16)) * (size_t)D_ + 2 * hi;

    v8f acc = {};
#pragma unroll
    for (int kk = 0; kk < D_; kk += 4) {
        v2f a = *(const v2f*)(qp + kk);
        v2f b = *(const v2f*)(kp + kk);
        // D = A(16x4) x B(4x16) + C ; emits v_wmma_f32_16x16x4_f32
        acc = __builtin_amdgcn_wmma_f32_16x16x4_f32(
            /*neg_a=*/false, a, /*neg_b=*/false, b,
            /*c_mod=*/(short)0, acc, /*reuse_a=*/false, /*reuse_b=*/false);
    }

    const float s = Msk[bp] * INV_SQRT_D;

    // C/D layout: VGPR r -> M = r + 8*hi, N = lo16. V and Out are streaming:
    // non-temporal so they don't thrash L2 (Q/K tiles want to stay resident).
    const size_t base = ((size_t)bp * L_ + m0) * (size_t)L_ + n0 + lo16;
#pragma unroll
    for (int r = 0; r < 8; ++r) {
        const size_t idx = base + (size_t)(r + 8 * hi) * L_;
        const float vv = __builtin_nontemporal_load(V + idx);
        __builtin_nontemporal_store(acc[r] * s * vv, Out + idx);
    }
}

extern "C" void kernel_launch(void* const* d_in, const int* in_sizes, int n_in,
                              void* d_out, int out_size, void* d_ws, size_t ws_size,
                              hipStream_t stream) {
    const float* Q   = (const float*)d_in[0];
    const float* Kt  = (const float*)d_in[1];
    const float* V   = (const float*)d_in[2];
    const float* Msk = (const float*)d_in[3];
    float* Out = (float*)d_out;

    // total wave-tiles = B*P * (L/16)^2 = 1024 * 256 = 262144 ; 8 waves/block
    const int blocks = (B_ * P_ * (L_ / 16) * (L_ / 16)) / 8; // 32768
    attn_qk_mask_v_kernel<<<blocks, 256, 0, stream>>>(Q, Kt, V, Msk, Out);
}